// AttentionMax_35338990911525
// MI455X (gfx1250) — compile-verified
//
#include <hip/hip_runtime.h>

typedef __attribute__((ext_vector_type(2))) float v2f;
typedef __attribute__((ext_vector_type(8))) float v8f;

#define N_SUPPORT 256
#define DIM       128

// One block per batch: 256 threads = 8 wave32.
// Each wave computes two 16-row tiles of correlation[b, :] in a single fused
// K-loop via V_WMMA_F32_16X16X4_F32 (query broadcast across B's 16 columns,
// B fragment shared between both tiles), then the block does a fused
// argmax + one-hot scatter.
__global__ __launch_bounds__(256)
void attnmax_wmma_kernel(const float* __restrict__ q,
                         const float* __restrict__ sub,
                         float* __restrict__ out) {
  __shared__ float lds_q[DIM];        // query vector for this batch
  __shared__ float corr[N_SUPPORT];   // correlations
  __shared__ float rv[N_SUPPORT];     // argmax reduction values
  __shared__ int   ri[N_SUPPORT];     // argmax reduction indices

  const int b    = blockIdx.x;
  const int tid  = threadIdx.x;
  const int wave = tid >> 5;   // 0..7 (wave32)
  const int lane = tid & 31;
  const int half = lane >> 4;  // 0: lanes 0-15, 1: lanes 16-31
  const int lrow = lane & 15;  // row within 16-row tile (A), col N (B/D)

  // Stage query into LDS once.
  if (tid < DIM) lds_q[tid] = q[(size_t)b * DIM + tid];
  __syncthreads();

  const float* subb = sub + (size_t)b * N_SUPPORT * DIM;

  // Two 16-row tiles per wave, fused into one K-loop sharing the B fragment.
  const int s0 = wave * 32;        // first tile's base support row
  const int s1 = wave * 32 + 16;   // second tile's base support row

  // A-matrix 16x4 layout: lane L -> row L%16, k = 2*(L/16) + {0,1}
  const float* arow0 = subb + (size_t)(s0 + lrow) * DIM + 2 * half;
  const float* arow1 = subb + (size_t)(s1 + lrow) * DIM + 2 * half;
  const float* qk    = lds_q + 2 * half;   // B row k = 2*half + r, same for all N

  v8f acc0 = {0.f, 0.f, 0.f, 0.f, 0.f, 0.f, 0.f, 0.f};
  v8f acc1 = {0.f, 0.f, 0.f, 0.f, 0.f, 0.f, 0.f, 0.f};

  #pragma unroll 4
  for (int k0 = 0; k0 < DIM; k0 += 4) {
    v2f bq = *(const v2f*)(qk + k0);          // 8B LDS load, shared by both tiles
    v2f a0 = *(const v2f*)(arow0 + k0);       // 8B aligned global loads
    v2f a1 = *(const v2f*)(arow1 + k0);
    // 8 args: (neg_a, A, neg_b, B, c_mod, C, reuse_a, reuse_b)
    acc0 = __builtin_amdgcn_wmma_f32_16x16x4_f32(
        false, a0, false, bq, (short)0, acc0, false, false);
    acc1 = __builtin_amdgcn_wmma_f32_16x16x4_f32(
        false, a1, false, bq, (short)0, acc1, false, false);
  }

  // D layout: VGPR r, lane L -> M = r + 8*(L/16), N = L%16.
  // All N columns are identical (query broadcast) -> column N==0 publishes.
  if (lrow == 0) {
    #pragma unroll
    for (int r = 0; r < 8; ++r) {
      corr[s0 + half * 8 + r] = acc0[r];
      corr[s1 + half * 8 + r] = acc1[r];
    }
  }
  __syncthreads();

  // Block-wide argmax over 256 correlations, first-index tie-break (jnp.argmax).
  rv[tid] = corr[tid];
  ri[tid] = tid;
  __syncthreads();
  #pragma unroll
  for (int s = N_SUPPORT / 2; s > 0; s >>= 1) {
    if (tid < s) {
      const float ov = rv[tid + s];
      const int   oi = ri[tid + s];
      const float mv = rv[tid];
      const int   mi = ri[tid];
      if (ov > mv || (ov == mv && oi < mi)) { rv[tid] = ov; ri[tid] = oi; }
    }
    __syncthreads();
  }
  const int best = ri[0];

  // One-hot scatter: out[b, s, 0] = (s == argmax)
  out[(size_t)b * N_SUPPORT + tid] = (tid == best) ? 1.0f : 0.0f;
}

extern "C" void kernel_launch(void* const* d_in, const int* in_sizes, int n_in,
                              void* d_out, int out_size, void* d_ws, size_t ws_size,
                              hipStream_t stream) {
  const float* q   = (const float*)d_in[0];  // [bz, 128]
  const float* sub = (const float*)d_in[1];  // [bz, 256, 128]
  float* out = (float*)d_out;                // [bz, 256, 1]

  const int bz = in_sizes[0] / DIM;          // 4096
  attnmax_wmma_kernel<<<bz, 256, 0, stream>>>(q, sub, out);
}